// LuongAttention_55353538511172
// MI455X (gfx1250) — compile-verified
//
#include <hip/hip_runtime.h>
#include <hip/hip_bf16.h>
#include <stdint.h>

// ---------------------------------------------------------------------------
// Luong attention, MI455X (gfx1250).
//   B=64, T=4096, U=256, D_DEC=256.
// Memory-bound: encoder_h = 256 MB read once (TDM -> LDS tiles), online softmax
// fuses score + context into a single HBM pass (~11 us at 23.3 TB/s).
// WMMA f32 16x16x4 used for the state GEMM and the score GEMV.
// ---------------------------------------------------------------------------

#define BB 64
#define TT_TOT 4096
#define UU 256
#define NS 4                   // splits of T per batch
#define TCHUNK (TT_TOT / NS)   // 1024 rows per workgroup
#define TROWS 64               // LDS tile rows
#define NTILES (TCHUNK / TROWS)
#define ROWF 258               // padded LDS row stride (floats): 256 + 2 (TDM pad)
#define BUF_FLOATS (TROWS * ROWF)

typedef float v2f __attribute__((ext_vector_type(2)));
typedef float v8f __attribute__((ext_vector_type(8)));
typedef unsigned int v4u __attribute__((ext_vector_type(4)));
typedef int v4i __attribute__((ext_vector_type(4)));
typedef int v8i __attribute__((ext_vector_type(8)));

// ---------------------------------------------------------------------------
// Kernel 1: state = decoder_s @ Wa_w + Wa_b   (64x256 <- 64x256 * 256x256)
// One wave per 16x16 output tile, V_WMMA_F32_16X16X4_F32, K=256 in steps of 4.
// A layout (16x4 f32): lanes 0-15 rows M=0..15 with K={0,1}; lanes 16-31 K={2,3}.
// B layout (4x16 f32): VGPR0 = K-row {0 | 2}, VGPR1 = K-row {1 | 3}, N = lane&15.
// C/D: VGPR r -> M = r (lanes 0-15) / M = r+8 (lanes 16-31), N = lane&15.
// ---------------------------------------------------------------------------
__global__ __launch_bounds__(32)
void state_gemm_kernel(const float* __restrict__ ds,
                       const float* __restrict__ Ww,
                       const float* __restrict__ Wb,
                       float* __restrict__ state) {
    const int tileM = (blockIdx.x >> 4) * 16;   // 4 M-tiles (B=64)
    const int tileN = (blockIdx.x & 15) * 16;   // 16 N-tiles (U=256)
    const int lane = threadIdx.x;
    const int lm = lane & 15;
    const int half = lane >> 4;

    v8f acc = {};
    const float* arow = ds + (size_t)(tileM + lm) * UU;  // decoder_s row
    #pragma unroll 4
    for (int k = 0; k < UU; k += 4) {
        v2f a, b;
        a.x = arow[k + half * 2];
        a.y = arow[k + half * 2 + 1];
        b.x = Ww[(size_t)(k + half * 2) * UU + tileN + lm];
        b.y = Ww[(size_t)(k + half * 2 + 1) * UU + tileN + lm];
        acc = __builtin_amdgcn_wmma_f32_16x16x4_f32(
            false, a, false, b, (short)0, acc, false, false);
    }
    const int col = tileN + lm;
    const float bias = Wb[col];
    #pragma unroll
    for (int r = 0; r < 8; ++r) {
        const int row = tileM + half * 8 + r;
        state[(size_t)row * UU + col] = acc[r] + bias;
    }
}

// ---------------------------------------------------------------------------
// TDM: DMA one 64x256 f32 tile of encoder_h into LDS, with 2-dword padding
// every 256 dwords so LDS rows are 258 floats (bank-conflict-free columns).
// D# group0: count=1, lds_addr, 57-bit global_addr, type=2.
// D# group1: data_size=4B, pad_enable, pad_interval=256dw, pad_amount=2dw,
//            tensor_dim0=256, tensor_dim1=64, tile_dim0=256, tile_dim1=64,
//            tensor_dim0_stride=256.
// Groups 2/3 unused (2D tile) -> zeros.
// ---------------------------------------------------------------------------
__device__ __forceinline__ void tdm_load_tile(const float* gptr, float* ldsptr) {
    const uint64_t ga = (uint64_t)(uintptr_t)gptr;
    // generic pointer to __shared__ has the workgroup LDS byte offset in [31:0]
    const uint32_t la = (uint32_t)(uintptr_t)(void*)ldsptr;
    v4u g0;
    g0.x = 1u;                                   // count = 1 descriptor
    g0.y = la;                                   // lds_addr
    g0.z = (uint32_t)ga;                         // global_addr[31:0]
    g0.w = (uint32_t)(ga >> 32) | 0x80000000u;   // global_addr[56:32] | type=2
    v8i g1;
    g1[0] = (2 << 16)        // data_size = 4 bytes
          | (1 << 20)        // pad_enable
          | (7 << 22)        // pad_interval: 256 dwords
          | (1 << 25);       // pad_amount: 2 dwords
    g1[1] = (int)(256u << 16);  // tensor_dim0 = 256 (low16 in bits[63:48])
    g1[2] = (int)(64u << 16);   // tensor_dim1 = 64  (low16 in bits[95:80])
    g1[3] = (int)(256u << 16);  // tile_dim0 = 256   (bits[127:112])
    g1[4] = 64;                 // tile_dim1 = 64    (bits[143:128])
    g1[5] = 256;                // tensor_dim0_stride = 256
    g1[6] = 0;
    g1[7] = 0;
    v4i z4 = {};
    v8i z8 = {};
    __builtin_amdgcn_tensor_load_to_lds(g0, g1, z4, z4, z8, 0);
}

// ---------------------------------------------------------------------------
// Kernel 2: fused score + online softmax + context, one HBM pass over H.
// Grid: B*NS workgroups of 256 threads (8 waves). Each WG owns (b, 1024 t's),
// streamed as 16 double-buffered 64-row LDS tiles loaded by the TDM.
// Waves 0-3 compute scores with WMMA (A = H tile rows from LDS, B = state in
// column 0); all 8 waves accumulate the rescaled context (1 u-column/thread).
// Raw masked scores stashed to d_ws for the final alpha pass.
// ---------------------------------------------------------------------------
__global__ __launch_bounds__(256)
void attn_stream_kernel(const float* __restrict__ H,
                        const int* __restrict__ maskp,
                        const float* __restrict__ state,
                        float* __restrict__ scores,
                        float* __restrict__ pms,
                        float* __restrict__ pctx) {
    extern __shared__ float smem[];
    float* bufs   = smem;                      // 2 * BUF_FLOATS
    float* stateL = smem + 2 * BUF_FLOATS;     // 256
    float* scT    = stateL + UU;               // 64 tile scores
    float* wT     = scT + TROWS;               // 64 tile exp-weights
    float* red    = wT + TROWS;                // [0]=tileMax [1]=sumw

    const int blk = blockIdx.x;
    const int b = blk / NS;
    const int chunk = blk % NS;
    const int tid = threadIdx.x;
    const int lane = tid & 31;
    const int wv = tid >> 5;

    stateL[tid] = state[(size_t)b * UU + tid];          // blockDim == U
    if (wv == 0)
        tdm_load_tile(H + ((size_t)b * TT_TOT + (size_t)chunk * TCHUNK) * UU, bufs);
    __syncthreads();

    float runM = -3.0e38f, runS = 0.0f, cacc = 0.0f;

    for (int tile = 0; tile < NTILES; ++tile) {
        const int t0 = chunk * TCHUNK + tile * TROWS;
        if (wv == 0) {
            if (tile + 1 < NTILES) {
                tdm_load_tile(H + ((size_t)b * TT_TOT + (size_t)(t0 + TROWS)) * UU,
                              bufs + ((tile + 1) & 1) * BUF_FLOATS);
                __builtin_amdgcn_s_wait_tensorcnt(1);   // current tile landed
            } else {
                __builtin_amdgcn_s_wait_tensorcnt(0);
            }
        }
        __syncthreads();
        float* buf = bufs + (tile & 1) * BUF_FLOATS;

        // ---- scores for 64 t-rows: waves 0-3, one 16-row m-tile each ----
        if (wv < 4) {
            const int lm = lane & 15;
            const int half = lane >> 4;
            v8f acc = {};
            const float* arow = buf + (wv * 16 + lm) * ROWF + half * 2;
            #pragma unroll 4
            for (int k = 0; k < UU; k += 4) {
                v2f a, bmat;
                a.x = arow[k];
                a.y = arow[k + 1];
                const float s0 = stateL[k + half * 2];
                const float s1 = stateL[k + half * 2 + 1];
                bmat.x = (lm == 0) ? s0 : 0.0f;   // state lives in column N=0
                bmat.y = (lm == 0) ? s1 : 0.0f;
                acc = __builtin_amdgcn_wmma_f32_16x16x4_f32(
                    false, a, false, bmat, (short)0, acc, false, false);
            }
            if (lm == 0) {                        // lanes 0 & 16 hold column 0
                float* dst = scT + wv * 16 + half * 8;
                #pragma unroll
                for (int r = 0; r < 8; ++r) dst[r] = acc[r];
            }
        }
        __syncthreads();

        // ---- mask, stash raw scores to scratch ----
        if (tid < TROWS) {
            const int t = t0 + tid;
            float sc = scT[tid];
            sc = (maskp[(size_t)b * TT_TOT + t] > 0) ? sc : -3.0e38f;
            scT[tid] = sc;
            scores[(size_t)b * TT_TOT + t] = sc;
        }
        __syncthreads();

        // ---- tile max ----
        if (wv == 0) {
            float v = fmaxf(scT[lane], scT[lane + 32]);
            #pragma unroll
            for (int off = 16; off; off >>= 1) v = fmaxf(v, __shfl_xor(v, off, 32));
            if (lane == 0) red[0] = v;
        }
        __syncthreads();

        const float newM = fmaxf(runM, red[0]);
        const float scale = __expf(runM - newM);   // 1.0 when both -3e38
        if (tid < TROWS) {
            const float sc = scT[tid];
            wT[tid] = (sc > -1.0e38f) ? __expf(sc - newM) : 0.0f;
        }
        __syncthreads();

        // ---- tile sum of weights ----
        if (wv == 0) {
            float v = wT[lane] + wT[lane + 32];
            #pragma unroll
            for (int off = 16; off; off >>= 1) v += __shfl_xor(v, off, 32);
            if (lane == 0) red[1] = v;
        }
        __syncthreads();

        runS = runS * scale + red[1];
        runM = newM;

        // ---- context accumulation: thread tid owns u-column tid ----
        float c = cacc * scale;
        const float* col = buf + tid;
        #pragma unroll 8
        for (int t = 0; t < TROWS; ++t) c += wT[t] * col[t * ROWF];
        cacc = c;
        __syncthreads();   // everyone done with buf before TDM reuses it
    }

    if (tid == 0) {
        pms[blk * 2] = runM;
        pms[blk * 2 + 1] = runS;
    }
    pctx[(size_t)blk * UU + tid] = cacc;
}

// ---------------------------------------------------------------------------
// Kernel 3: combine NS split-partials per batch, emit ct_sum and alpha.
// ---------------------------------------------------------------------------
__global__ __launch_bounds__(256)
void attn_combine_kernel(const float* __restrict__ scores,
                         const float* __restrict__ pms,
                         const float* __restrict__ pctx,
                         float* __restrict__ out) {
    const int b = blockIdx.x;
    const int tid = threadIdx.x;

    float mj[NS], sj[NS];
    float m = -3.0e38f;
    #pragma unroll
    for (int j = 0; j < NS; ++j) {
        mj[j] = pms[(b * NS + j) * 2];
        sj[j] = pms[(b * NS + j) * 2 + 1];
        m = fmaxf(m, mj[j]);
    }
    float s = 0.0f, coef[NS];
    #pragma unroll
    for (int j = 0; j < NS; ++j) {
        coef[j] = (mj[j] > -1.0e38f) ? __expf(mj[j] - m) : 0.0f;
        s += sj[j] * coef[j];
    }
    const float inv_s = 1.0f / s;   // >=1024 valid tokens -> s > 0

    float ct = 0.0f;
    #pragma unroll
    for (int j = 0; j < NS; ++j)
        ct += coef[j] * pctx[((size_t)(b * NS + j)) * UU + tid];
    out[(size_t)b * UU + tid] = ct * inv_s;

    float* alpha = out + (size_t)BB * UU;
    #pragma unroll 2
    for (int k = 0; k < TT_TOT / 256; ++k) {
        const int t = k * 256 + tid;
        const float sc = scores[(size_t)b * TT_TOT + t];
        alpha[(size_t)b * TT_TOT + t] =
            (sc > -1.0e38f) ? __expf(sc - m) * inv_s : 0.0f;
    }
}

// ---------------------------------------------------------------------------
// launch: inputs = {encoder_h, decoder_s, mask, Wa_w, Wa_b}
// d_out = [ct_sum (B*U) | alpha (B*T)]
// d_ws  = [state (B*U) | scores (B*T) | pms (B*NS*2) | pctx (B*NS*U)] ~1.4 MB
// ---------------------------------------------------------------------------
extern "C" void kernel_launch(void* const* d_in, const int* in_sizes, int n_in,
                              void* d_out, int out_size, void* d_ws, size_t ws_size,
                              hipStream_t stream) {
    const float* H  = (const float*)d_in[0];
    const float* ds = (const float*)d_in[1];
    const int* mask = (const int*)d_in[2];
    const float* Ww = (const float*)d_in[3];
    const float* Wb = (const float*)d_in[4];
    float* out = (float*)d_out;

    float* state  = (float*)d_ws;
    float* scores = state + BB * UU;
    float* pms    = scores + (size_t)BB * TT_TOT;
    float* pctx   = pms + BB * NS * 2;

    const size_t smem_bytes =
        (size_t)(2 * BUF_FLOATS + UU + TROWS + TROWS + 2) * sizeof(float);
    // ~131 KB dynamic LDS (< 320 KB/WGP): allow it explicitly.
    (void)hipFuncSetAttribute((const void*)attn_stream_kernel,
                              hipFuncAttributeMaxDynamicSharedMemorySize,
                              (int)smem_bytes);

    state_gemm_kernel<<<(BB / 16) * (UU / 16), 32, 0, stream>>>(ds, Ww, Wb, state);
    attn_stream_kernel<<<BB * NS, 256, smem_bytes, stream>>>(H, mask, state,
                                                             scores, pms, pctx);
    attn_combine_kernel<<<BB, 256, 0, stream>>>(scores, pms, pctx, out);
}